// MyDTI_78262894068335
// MI455X (gfx1250) — compile-verified
//
#include <hip/hip_runtime.h>

typedef __attribute__((ext_vector_type(2))) float v2f;
typedef __attribute__((ext_vector_type(8))) float v8f;

#define DDIM    128
#define RREL    3
#define NNODES  100000
#define NEDGES  1600000

// ---------------------------------------------------------------------------
// GEMM kernel: for one layer, one wave computes a 16-row M-tile for one weight
// matrix (r = 0..2 relation, r = 3 self-loop) across the full 128-col output.
//   r < 3 : xW[n, r, :]  = H[n, :] @ Wrel[r]          (messages, pre-transform)
//   r = 3 : out[n, :]    = H[n, :] @ Wself + bias     (scatter-add init)
// Uses V_WMMA_F32_16X16X4_F32 (exact fp32, matches reference numerics).
// ---------------------------------------------------------------------------
__global__ __launch_bounds__(128) void rgcn_gemm_wmma(
    const float* __restrict__ H,       // [N][128]
    const float* __restrict__ Wrel,    // [3][128][128]  (i, o)
    const float* __restrict__ Wself,   // [128][128]
    const float* __restrict__ bias,    // [128]
    float* __restrict__ xW,            // [N][3][128]
    float* __restrict__ out)           // [N][128]
{
  const int wave  = blockIdx.x * (blockDim.x >> 5) + (threadIdx.x >> 5);
  const int lane  = threadIdx.x & 31;
  const int mtile = wave >> 2;              // 6250 M-tiles of 16 rows
  const int r     = wave & 3;               // 0..2 = relation, 3 = self-loop
  if (mtile >= NNODES / 16) return;

  const float* Wmat = (r < RREL) ? (Wrel + (size_t)r * DDIM * DDIM) : Wself;

  const int row0  = mtile * 16;
  const int m     = lane & 15;              // M index (A) / N index (B,C)
  const int khalf = lane >> 4;              // lanes 16-31 hold K = k0+2, k0+3

  v8f c[8];
  #pragma unroll
  for (int t = 0; t < 8; ++t)
    c[t] = (v8f){0.f, 0.f, 0.f, 0.f, 0.f, 0.f, 0.f, 0.f};

  const float* hrow = H + (size_t)(row0 + m) * DDIM;

  #pragma unroll 4
  for (int kk = 0; kk < DDIM / 4; ++kk) {
    const int k0 = kk * 4 + 2 * khalf;
    // A fragment (16x4 f32): lane holds A[m][k0], A[m][k0+1]
    v2f a = *(const v2f*)(hrow + k0);
    const float* wk0 = Wmat + (size_t)k0 * DDIM;
    #pragma unroll
    for (int t = 0; t < 8; ++t) {
      // B fragment (4x16 f32): lane holds B[k0][col], B[k0+1][col]
      v2f b;
      b.x = wk0[t * 16 + m];
      b.y = wk0[DDIM + t * 16 + m];
      c[t] = __builtin_amdgcn_wmma_f32_16x16x4_f32(
          /*neg_a=*/false, a, /*neg_b=*/false, b,
          /*c_mod=*/(short)0, c[t], /*reuse_a=*/false, /*reuse_b=*/false);
    }
  }

  // C/D layout: VGPR v -> M=v (lanes 0-15) / M=8+v (lanes 16-31), N = lane&15
  #pragma unroll
  for (int t = 0; t < 8; ++t) {
    const int col = t * 16 + m;
    #pragma unroll
    for (int v = 0; v < 8; ++v) {
      const int row = row0 + v + 8 * khalf;
      if (r < RREL) {
        xW[((size_t)row * RREL + r) * DDIM + col] = c[t][v];
      } else {
        out[(size_t)row * DDIM + col] = c[t][v] + bias[col];
      }
    }
  }
}

// ---------------------------------------------------------------------------
// Edge kernel: one wave per edge; 32 lanes x float4 cover the 128-float
// message. Scatter-add resolves in L2 (dest buffer = 51 MB < 192 MB L2).
// ---------------------------------------------------------------------------
__global__ __launch_bounds__(256) void rgcn_edge_scatter(
    const float* __restrict__ xW,      // [N][3][128]
    const int* __restrict__ src,
    const int* __restrict__ dst,
    const int* __restrict__ et,
    float* __restrict__ out)           // [N][128]
{
  const int e    = blockIdx.x * (blockDim.x >> 5) + (threadIdx.x >> 5);
  const int lane = threadIdx.x & 31;
  if (e >= NEDGES) return;

  const int s = src[e];
  const int t = dst[e];
  const int r = et[e];

  const float4 msg = *(const float4*)(xW + ((size_t)s * RREL + r) * DDIM + lane * 4);
  float* o = out + (size_t)t * DDIM + lane * 4;
  atomicAdd(o + 0, msg.x);
  atomicAdd(o + 1, msg.y);
  atomicAdd(o + 2, msg.z);
  atomicAdd(o + 3, msg.w);
}

// ---------------------------------------------------------------------------
// Host driver: 3 layers, ping-pong h through workspace, final layer -> d_out.
// Workspace: xW (153.6 MB) + hA (51.2 MB) + hB (51.2 MB) = 256 MB.
// ---------------------------------------------------------------------------
extern "C" void kernel_launch(void* const* d_in, const int* in_sizes, int n_in,
                              void* d_out, int out_size, void* d_ws, size_t ws_size,
                              hipStream_t stream) {
  const float* features = (const float*)d_in[0];   // [N,128]
  const float* W        = (const float*)d_in[1];   // [3,3,128,128]
  const float* Wself    = (const float*)d_in[2];   // [3,128,128]
  const float* bias     = (const float*)d_in[3];   // [3,128]
  const int*   src      = (const int*)d_in[4];
  const int*   dst      = (const int*)d_in[5];
  const int*   et       = (const int*)d_in[6];
  float*       out      = (float*)d_out;

  float* xW = (float*)d_ws;                              // N*3*128 floats
  float* hA = xW + (size_t)NNODES * RREL * DDIM;         // N*128
  float* hB = hA + (size_t)NNODES * DDIM;                // N*128

  const int gemm_waves = (NNODES / 16) * 4;              // 25000 waves
  dim3 gemm_grid((gemm_waves + 3) / 4);                  // 4 waves / block
  dim3 gemm_block(128);
  dim3 sc_grid((NEDGES + 7) / 8);                        // 8 edges / block
  dim3 sc_block(256);

  const float* hin = features;
  float* houts[3] = {hA, hB, out};
  for (int l = 0; l < 3; ++l) {
    rgcn_gemm_wmma<<<gemm_grid, gemm_block, 0, stream>>>(
        hin,
        W + (size_t)l * RREL * DDIM * DDIM,
        Wself + (size_t)l * DDIM * DDIM,
        bias + (size_t)l * DDIM,
        xW, houts[l]);
    rgcn_edge_scatter<<<sc_grid, sc_block, 0, stream>>>(
        xW, src, dst, et, houts[l]);
    hin = houts[l];
  }
}